// zhushuilayer_50629074486006
// MI455X (gfx1250) — compile-verified
//
#include <hip/hip_runtime.h>
#include <math.h>

// ---------------------------------------------------------------------------
// zhushui layer: out = outputs + noise * sqrt(waterfill_power(var_per_sample))
// B=64 samples, 256*56*56 = 802816 f32 elements per sample.
// Memory-bound: ~822MB traffic -> ~35us roofline @ 23.3 TB/s.
// ---------------------------------------------------------------------------

typedef __attribute__((ext_vector_type(2))) float v2f;
typedef __attribute__((ext_vector_type(4))) float v4f;
typedef __attribute__((ext_vector_type(8))) float v8f;

#define NB       64        // batch (samples)
#define SAMP_V4  200704    // float4 per sample (802816/4)
#define BPS      98        // blocks per sample
#define TPB      256       // threads per block (8 waves on wave32)
#define TPS      (BPS*TPB) // 25088 threads per sample
#define V4PT     8         // float4 per thread (25088*8 = 200704 exactly)
#define NELEM_F  802816.0f
#define KFRAC    0.2f

// Full 32-lane wave sum using the f32 WMMA as a reduction engine.
// A (16x4 f32, documented layout): lane m (0-15) supplies K0/K1, lane m+16
// supplies K2/K3. With a = (p, 0) and B = ones, D[m][n] = p[m] + p[m+16].
// Lanes 0-15 hold rows 0-7 in c[0..7], lanes 16-31 hold rows 8-15, so the
// per-lane sum of c[0..7] gives half the wave; one xor-16 shuffle finishes.
__device__ __forceinline__ float wave_sum_wmma(float p) {
  v2f a; a.x = p;   a.y = 0.0f;
  v2f b; b.x = 1.0f; b.y = 1.0f;
  v8f c = {};
  c = __builtin_amdgcn_wmma_f32_16x16x4_f32(false, a, false, b,
                                            (short)0, c, false, false);
  float t = ((c[0] + c[1]) + (c[2] + c[3])) + ((c[4] + c[5]) + (c[6] + c[7]));
  t += __shfl_xor(t, 16, 32);
  return t;  // full 32-lane sum, replicated in every lane
}

// Pass 1: per-block (sum, sumsq) partials. Fully coalesced b128 loads,
// per-thread f32 accumulation, WMMA wave reduction, LDS block combine.
__global__ void zs_reduce_kernel(const float* __restrict__ outputs,
                                 float* __restrict__ pairs) {
  const int b = blockIdx.y;
  const int t = blockIdx.x * TPB + threadIdx.x;            // 0..TPS-1
  const v4f* o4 = (const v4f*)outputs + (size_t)b * SAMP_V4;

  float psum = 0.0f, psq = 0.0f;
#pragma unroll
  for (int i = 0; i < V4PT; ++i) {
    v4f x = o4[t + i * TPS];
    psum += (x.x + x.y) + (x.z + x.w);
    psq = fmaf(x.x, x.x, psq);
    psq = fmaf(x.y, x.y, psq);
    psq = fmaf(x.z, x.z, psq);
    psq = fmaf(x.w, x.w, psq);
  }

  const float wsum = wave_sum_wmma(psum);
  const float wsq  = wave_sum_wmma(psq);

  __shared__ float lsum[8], lsq[8];
  const int lane = threadIdx.x & 31;
  const int wv   = threadIdx.x >> 5;
  if (lane == 0) { lsum[wv] = wsum; lsq[wv] = wsq; }
  __syncthreads();
  if (threadIdx.x == 0) {
    float bs = 0.0f, bq = 0.0f;
#pragma unroll
    for (int w = 0; w < 8; ++w) { bs += lsum[w]; bq += lsq[w]; }
    v2f pr; pr.x = bs; pr.y = bq;
    ((v2f*)pairs)[b * BPS + blockIdx.x] = pr;
  }
}

// Pass 2: one block of 64 threads. Deterministic final accumulation of the
// 98 block partials per sample, unbiased variance, parallel water-fill
// (stable rank sort + cumsum, all O(64^2) in LDS — negligible).
__global__ void zs_waterfill_kernel(const float* __restrict__ pairs,
                                    float* __restrict__ scale) {
  __shared__ float v[NB], svals[NB], csum[NB];
  __shared__ int   flags[NB];
  const int t = threadIdx.x;  // sample id, 0..63

  double ds = 0.0, dq = 0.0;
  const v2f* p2 = (const v2f*)pairs + t * BPS;
  for (int i = 0; i < BPS; ++i) { v2f pr = p2[i]; ds += (double)pr.x; dq += (double)pr.y; }
  const float sum = (float)ds, sq = (float)dq;
  const float var = (sq - sum * sum / NELEM_F) / (NELEM_F - 1.0f);  // ddof=1
  v[t] = var;
  __syncthreads();

  // stable ascending rank (ties broken by index) == jnp.sort
  int rank = 0;
  for (int j = 0; j < NB; ++j) {
    const float vj = v[j];
    rank += (vj < var) || (vj == var && j < t);
  }
  svals[rank] = var;
  __syncthreads();

  float cs = 0.0f;
  for (int j = 0; j <= t; ++j) cs += svals[j];   // inclusive cumsum of sorted
  csum[t] = cs;
  float total = 0.0f;
  for (int j = 0; j < NB; ++j) total += v[j];
  total *= KFRAC;

  flags[t] = (((float)(t + 1) * svals[t] - cs) < total) ? 1 : 0;
  __syncthreads();

  int cnt = 0;
  for (int j = 0; j < NB; ++j) cnt += flags[j];
  if (cnt < 1) cnt = 1;
  const float level = (total + csum[cnt - 1]) / (float)cnt;
  const float power = fmaxf(level - var, 0.0f);
  scale[t] = sqrtf(power);
}

// Pass 3: out = outputs + noise * scale[b]. b128 vector traffic; `noise` and
// `out` are streamed once -> non-temporal, keeping L2 (192MB) for `outputs`
// which was just read by pass 1.
__global__ void zs_apply_kernel(const float* __restrict__ outputs,
                                const float* __restrict__ noise,
                                const float* __restrict__ scale,
                                float* __restrict__ out) {
  const int b = blockIdx.y;
  const int t = blockIdx.x * TPB + threadIdx.x;
  const float s = scale[b];
  const size_t base = (size_t)b * SAMP_V4;
  const v4f* o4 = (const v4f*)outputs + base;
  const v4f* n4 = (const v4f*)noise + base;
  v4f*       d4 = (v4f*)out + base;
#pragma unroll
  for (int i = 0; i < V4PT; ++i) {
    const int idx = t + i * TPS;
    v4f o  = o4[idx];
    v4f nz = __builtin_nontemporal_load(&n4[idx]);
    v4f r  = o + nz * s;
    __builtin_nontemporal_store(r, &d4[idx]);
  }
}

extern "C" void kernel_launch(void* const* d_in, const int* in_sizes, int n_in,
                              void* d_out, int out_size, void* d_ws, size_t ws_size,
                              hipStream_t stream) {
  (void)in_sizes; (void)n_in; (void)out_size; (void)ws_size;
  const float* outputs = (const float*)d_in[0];
  const float* noise   = (const float*)d_in[1];
  float* out   = (float*)d_out;
  float* pairs = (float*)d_ws;                 // NB*BPS*2 floats (~50KB)
  float* scale = pairs + NB * BPS * 2;         // NB floats

  dim3 grid(BPS, NB);
  zs_reduce_kernel<<<grid, TPB, 0, stream>>>(outputs, pairs);
  zs_waterfill_kernel<<<1, NB, 0, stream>>>(pairs, scale);
  zs_apply_kernel<<<grid, TPB, 0, stream>>>(outputs, noise, scale, out);
}